// AlignBlock_69020124447124
// MI455X (gfx1250) — compile-verified
//
#include <hip/hip_runtime.h>
#include <hip/hip_bf16.h>
#include <math.h>

// Problem constants (from reference setup_inputs)
#define NB 2      // batch
#define NH 16     // hidden channels
#define NC 4      // mic/ref channels
#define NT 400    // time
#define NF 129    // freq
#define NFP 132   // freq padded to multiple of 4 (WMMA K-step)
#define ND 100    // delay window
#define KPAD 112  // zero rows in front of each K plane (max negative band reach)
#define NTP (NT + KPAD)  // 512 padded time rows for K

typedef __attribute__((ext_vector_type(2))) float v2f;
typedef __attribute__((ext_vector_type(8))) float v8f;

// ---------------------------------------------------------------------------
// Kernel 1: 1x1 channel mix.
//   Qp: (B,H,T,Fp)          fp32, zero for f>=129
//   Kp: (B,H,KPAD+T,Fp)     fp32, rows t<KPAD are zeros (causal pad as data,
//                           so the WMMA kernel needs no per-lane selects)
// ---------------------------------------------------------------------------
__global__ void prep_qk_kernel(const float* __restrict__ xm, const float* __restrict__ xr,
                               const float* __restrict__ Wm, const float* __restrict__ bm,
                               const float* __restrict__ Wr, const float* __restrict__ br,
                               float* __restrict__ Qp, float* __restrict__ Kp) {
    int idx = blockIdx.x * blockDim.x + threadIdx.x;
    const int total = NB * NH * NTP * NFP;
    if (idx >= total) return;
    int fp = idx % NFP;
    int tp = (idx / NFP) % NTP;
    int h  = (idx / (NFP * NTP)) % NH;
    int b  =  idx / (NFP * NTP * NH);
    int t  = tp - KPAD;

    float q = 0.f, k = 0.f;
    if (t >= 0 && fp < NF) {
        q = bm[h];
        k = br[h];
        #pragma unroll
        for (int c = 0; c < NC; ++c) {
            int xo = ((b * NC + c) * NT + t) * NF + fp;
            q = fmaf(Wm[h * NC + c], xm[xo], q);
            k = fmaf(Wr[h * NC + c], xr[xo], k);
        }
    }
    Kp[idx] = k;                                   // includes zero pad rows
    if (t >= 0)
        Qp[(size_t)((b * NH + h) * NT + t) * NFP + fp] = q;
}

// ---------------------------------------------------------------------------
// Kernel 2: banded Q*K^T via V_WMMA_F32_16X16X4_F32.
// One wave per (bh, t-tile, j). u0 = t0 - 112 + 16*j covers the causal band
// d = u - t + 99 in [0,100); rows u<0 read the zero pad in Kp directly.
// 6400 waves, 33 chained fp32 WMMAs each (K = 132); inner loop is pure
// load/load/wmma — no selects in front of the matrix pipe.
// ---------------------------------------------------------------------------
__global__ void __launch_bounds__(256) vcorr_wmma_kernel(const float* __restrict__ Qp,
                                                         const float* __restrict__ Kp,
                                                         float* __restrict__ V) {
    int wid  = (blockIdx.x * blockDim.x + threadIdx.x) >> 5; // global wave id, 0..6399
    int lane = threadIdx.x & 31;

    int j  = wid & 7;          // u-tile within band
    int it = (wid >> 3) % 25;  // t-tile
    int bh = wid / 200;        // fused (b,h), 0..31

    int t0 = it * 16;
    int u0 = t0 - 112 + 16 * j;

    int row = lane & 15;        // M (for A) / N (for B) row within tile
    int kb  = (lane >> 4) * 2;  // K pair: lanes 0-15 -> K{0,1}, lanes 16-31 -> K{2,3}

    const float* qbase = Qp + (size_t)bh * NT  * NFP + (size_t)(t0 + row)          * NFP + kb;
    const float* kbase = Kp + (size_t)bh * NTP * NFP + (size_t)(KPAD + u0 + row)   * NFP + kb;
    // KPAD + u0 + row in [0, 511] always: in-bounds, zeros where u < 0.

    v8f acc = {0.f, 0.f, 0.f, 0.f, 0.f, 0.f, 0.f, 0.f};

    for (int i = 0; i < NFP / 4; ++i) {
        v2f a = *(const v2f*)(qbase + 4 * i);
        v2f b = *(const v2f*)(kbase + 4 * i);
        // D = A(16x4 f32) * B(4x16 f32) + C  (exact fp32 on the matrix pipe)
        acc = __builtin_amdgcn_wmma_f32_16x16x4_f32(
            /*neg_a=*/false, a, /*neg_b=*/false, b,
            /*c_mod=*/(short)0, acc, /*reuse_a=*/false, /*reuse_b=*/false);
    }

    // Scatter C/D tile: VGPR r -> M = r + 8*(lane>=16), N = lane&15.
    int n     = row;
    int mbase = (lane >> 4) * 8;
    float* Vb = V + (size_t)bh * NT * ND;
    #pragma unroll
    for (int r = 0; r < 8; ++r) {
        int m = r + mbase;
        int t = t0 + m;
        int d = 16 * j + n - m - 13;    // d = (u0+n) - (t0+m) + 99
        if (d >= 0 && d < ND)
            Vb[(size_t)t * ND + d] = acc[r];
    }
}

// ---------------------------------------------------------------------------
// Kernel 3: per (b,t) workgroup (128 threads):
//   conv (16->1, 5x3, causal T-pad, D-pad 1/1) -> softmax over D ->
//   out[b,c,t,f] = sum_d x_ref[b,c,t-99+d,f] * A[d]
// ---------------------------------------------------------------------------
__global__ void __launch_bounds__(128) conv_softmax_apply_kernel(
        const float* __restrict__ V, const float* __restrict__ Wc,
        const float* __restrict__ bc, const float* __restrict__ xr,
        float* __restrict__ out) {
    int bt  = blockIdx.x;
    int b   = bt / NT;
    int t   = bt % NT;
    int tid = threadIdx.x;

    __shared__ float sA[ND];
    __shared__ float red[128];

    float vconv = 0.f;
    float vpad  = -3.0e38f; // -inf pad for max reduction
    if (tid < ND) {
        int d = tid;
        vconv = bc[0];
        for (int h = 0; h < NH; ++h) {
            const float* Vbh = V + (size_t)((b * NH + h) * NT) * ND;
            #pragma unroll
            for (int kt = 0; kt < 5; ++kt) {
                int tt = t - 4 + kt;
                if (tt < 0) continue;            // causal zero pad on time
                const float* Vrow = Vbh + (size_t)tt * ND;
                const float* w    = Wc + h * 15 + kt * 3;
                float s = Vrow[d] * w[1];
                if (d >= 1)      s = fmaf(Vrow[d - 1], w[0], s);
                if (d <= ND - 2) s = fmaf(Vrow[d + 1], w[2], s);
                vconv += s;
            }
        }
        vpad = vconv;
    }

    // max reduce over 128 (tail padded with -inf)
    red[tid] = vpad;
    __syncthreads();
    for (int s = 64; s > 0; s >>= 1) {
        if (tid < s) red[tid] = fmaxf(red[tid], red[tid + s]);
        __syncthreads();
    }
    float mx = red[0];
    __syncthreads();

    float e = (tid < ND) ? __expf(vconv - mx) : 0.f;
    red[tid] = e;
    __syncthreads();
    for (int s = 64; s > 0; s >>= 1) {
        if (tid < s) red[tid] += red[tid + s];
        __syncthreads();
    }
    float inv = 1.f / red[0];
    __syncthreads();
    if (tid < ND) sA[tid] = e * inv;
    __syncthreads();

    // apply attention over delays to x_ref
    int dmin = (t >= ND - 1) ? 0 : (ND - 1 - t);  // zero-padded region skipped
    for (int idx = tid; idx < NC * NF; idx += 128) {
        int c = idx / NF;
        int f = idx % NF;
        const float* col = xr + (size_t)((b * NC + c) * NT) * NF + f;
        float a0 = 0.f;
        for (int d = dmin; d < ND; ++d) {
            int u = t - (ND - 1) + d;             // >= 0 by dmin
            a0 = fmaf(col[(size_t)u * NF], sA[d], a0);
        }
        out[(size_t)((b * NC + c) * NT + t) * NF + f] = a0;
    }
}

// ---------------------------------------------------------------------------
extern "C" void kernel_launch(void* const* d_in, const int* in_sizes, int n_in,
                              void* d_out, int out_size, void* d_ws, size_t ws_size,
                              hipStream_t stream) {
    const float* x_mic  = (const float*)d_in[0];
    const float* x_ref  = (const float*)d_in[1];
    const float* W_mic  = (const float*)d_in[2];
    const float* b_mic  = (const float*)d_in[3];
    const float* W_ref  = (const float*)d_in[4];
    const float* b_ref  = (const float*)d_in[5];
    const float* W_conv = (const float*)d_in[6];
    const float* b_conv = (const float*)d_in[7];
    float* out = (float*)d_out;

    // Workspace layout (fp32): Qp | Kp(padded) | V   (~20.5 MB total)
    float* ws = (float*)d_ws;
    const size_t q_elems = (size_t)NB * NH * NT  * NFP;   // 1,689,600
    const size_t k_elems = (size_t)NB * NH * NTP * NFP;   // 2,162,688
    float* Qp = ws;
    float* Kp = ws + q_elems;
    float* V  = ws + q_elems + k_elems;                   // NB*NH*NT*ND

    // 1) channel mix into padded Q/K (K carries the causal zero rows)
    int total = NB * NH * NTP * NFP;
    prep_qk_kernel<<<(total + 255) / 256, 256, 0, stream>>>(
        x_mic, x_ref, W_mic, b_mic, W_ref, b_ref, Qp, Kp);

    // 2) banded QK^T on the matrix pipe: 6400 waves -> 800 blocks x 8 waves
    vcorr_wmma_kernel<<<800, 256, 0, stream>>>(Qp, Kp, V);

    // 3) conv + softmax + delay-weighted sum, one workgroup per (b,t)
    conv_softmax_apply_kernel<<<NB * NT, 128, 0, stream>>>(
        V, W_conv, b_conv, x_ref, out);
}